// GCN_29076928594465
// MI455X (gfx1250) — compile-verified
//
#include <hip/hip_runtime.h>
#include <hip/hip_bf16.h>

typedef __attribute__((ext_vector_type(2))) float v2f;
typedef __attribute__((ext_vector_type(8))) float v8f;

#define IN_DIM  512
#define HID_DIM 64
#define OUT_DIM 32

// ---------------------------------------------------------------------------
// 0. init: zero layer-1 accumulator + final output accumulator, deg = 1.0
//    (self-loop contribution to the degree is folded into the init value)
// ---------------------------------------------------------------------------
__global__ void gcn_init(float* __restrict__ deg, float* __restrict__ agg1,
                         float* __restrict__ out2, int n) {
    int total = n * HID_DIM;
    for (int i = blockIdx.x * blockDim.x + threadIdx.x; i < total;
         i += gridDim.x * blockDim.x) {
        agg1[i] = 0.0f;
        if (i < n * OUT_DIM) out2[i] = 0.0f;
        if (i < n)           deg[i]  = 1.0f;   // self loop
    }
}

// ---------------------------------------------------------------------------
// 1. degree: deg[col[e]] += 1  (L2-resident float atomics)
// ---------------------------------------------------------------------------
__global__ void gcn_degree(const int* __restrict__ col, int E,
                           float* __restrict__ deg) {
    int i = blockIdx.x * blockDim.x + threadIdx.x;
    if (i < E) atomicAdd(&deg[col[i]], 1.0f);
}

// ---------------------------------------------------------------------------
// 2. dinv = rsqrt(deg) in place
// ---------------------------------------------------------------------------
__global__ void gcn_dinv(float* __restrict__ deg, int n) {
    int i = blockIdx.x * blockDim.x + threadIdx.x;
    if (i < n) {
        float d = deg[i];
        deg[i] = (d > 0.0f) ? rsqrtf(d) : 0.0f;
    }
}

// ---------------------------------------------------------------------------
// 3. GEMM1: H[100000,64] = X[100000,512] @ W[512,64] using V_WMMA_F32_16X16X4
//    block = 256 threads = 8 waves; each wave owns a 16-row x 64-col stripe
//    (4 WMMA tiles, A fragment reused across the 4 B tiles).
//    W chunk (K=128) held transposed in LDS: sW[n*132 + k], stride 132 ==
//    4 mod 64 banks -> all 32 lanes of a b64 B-fragment load hit distinct
//    banks; even k offsets keep 8-byte alignment.
// ---------------------------------------------------------------------------
__global__ __launch_bounds__(256) void gcn_gemm1(const float* __restrict__ X,
                                                 const float* __restrict__ W,
                                                 float* __restrict__ H,
                                                 int nRows) {
    __shared__ float sW[HID_DIM * 132];          // ~33.8 KB
    const int lane = threadIdx.x & 31;
    const int wave = threadIdx.x >> 5;
    const int rowBase = (blockIdx.x * 8 + wave) * 16;
    const bool active = rowBase < nRows;         // wave-uniform: EXEC stays full
    const int ln = lane & 15;                    // column / M-row within tile
    const int kh = lane >> 4;                    // K-half select (ISA A layout)

    v8f acc0 = {}, acc1 = {}, acc2 = {}, acc3 = {};
    const float* xrow = X + (size_t)(rowBase + ln) * IN_DIM;

    for (int kc = 0; kc < IN_DIM; kc += 128) {
        // cooperative transpose-load of the W chunk (coalesced global reads)
        for (int i = threadIdx.x; i < HID_DIM * 128; i += 256) {
            int k  = i >> 6;      // 0..127
            int nn = i & 63;      // 0..63
            sW[nn * 132 + k] = W[(size_t)(kc + k) * HID_DIM + nn];
        }
        __syncthreads();

        if (active) {
#pragma unroll 4
            for (int k0 = 0; k0 < 128; k0 += 4) {
                const int kb = k0 + kh * 2;                   // even -> 8B align
                float2 av = *(const float2*)(xrow + kc + kb); // A: 16x4 f32 frag
                v2f a; a[0] = av.x; a[1] = av.y;

                const float* bp = &sW[ln * 132 + kb];
                float2 w0 = *(const float2*)(bp);
                float2 w1 = *(const float2*)(bp + 16 * 132);
                float2 w2 = *(const float2*)(bp + 32 * 132);
                float2 w3 = *(const float2*)(bp + 48 * 132);

                v2f b;
                b[0] = w0.x; b[1] = w0.y;
                acc0 = __builtin_amdgcn_wmma_f32_16x16x4_f32(false, a, false, b,
                                                             (short)0, acc0, false, false);
                b[0] = w1.x; b[1] = w1.y;
                acc1 = __builtin_amdgcn_wmma_f32_16x16x4_f32(false, a, false, b,
                                                             (short)0, acc1, false, false);
                b[0] = w2.x; b[1] = w2.y;
                acc2 = __builtin_amdgcn_wmma_f32_16x16x4_f32(false, a, false, b,
                                                             (short)0, acc2, false, false);
                b[0] = w3.x; b[1] = w3.y;
                acc3 = __builtin_amdgcn_wmma_f32_16x16x4_f32(false, a, false, b,
                                                             (short)0, acc3, false, false);
            }
        }
        __syncthreads();
    }

    if (active) {
        // C/D layout: VGPR j holds row (j + 8*kh), column = ln (+16 per tile)
        float* hp = H + (size_t)(rowBase + kh * 8) * HID_DIM + ln;
#pragma unroll
        for (int j = 0; j < 8; ++j) {
            hp[(size_t)j * HID_DIM +  0] = acc0[j];
            hp[(size_t)j * HID_DIM + 16] = acc1[j];
            hp[(size_t)j * HID_DIM + 32] = acc2[j];
            hp[(size_t)j * HID_DIM + 48] = acc3[j];
        }
    }
}

// ---------------------------------------------------------------------------
// 4. scatter: agg[col] += H[row] * dinv[row]*dinv[col] over E edges plus
//    N self-loops (slots E..E+N-1). float4 gathers, f32 atomics (L2).
// ---------------------------------------------------------------------------
template <int F>
__global__ void gcn_scatter(const int* __restrict__ row,
                            const int* __restrict__ col,
                            const float* __restrict__ dinv,
                            const float* __restrict__ H,
                            float* __restrict__ agg,
                            int E, int n) {
    int total = E + n;
    for (int i = blockIdx.x * blockDim.x + threadIdx.x; i < total;
         i += gridDim.x * blockDim.x) {
        int r, c;
        if (i < E) { r = row[i]; c = col[i]; }
        else       { r = c = i - E; }                 // self loop
        float norm = dinv[r] * dinv[c];
        const float4* hp = (const float4*)(H + (size_t)r * F);
        float* ap = agg + (size_t)c * F;
#pragma unroll
        for (int f = 0; f < F / 4; ++f) {
            float4 h = hp[f];
            atomicAdd(ap + f * 4 + 0, h.x * norm);
            atomicAdd(ap + f * 4 + 1, h.y * norm);
            atomicAdd(ap + f * 4 + 2, h.z * norm);
            atomicAdd(ap + f * 4 + 3, h.w * norm);
        }
    }
}

// ---------------------------------------------------------------------------
// 5. bias + ReLU (layer 1 activation)
// ---------------------------------------------------------------------------
__global__ void gcn_bias_relu(const float* __restrict__ agg,
                              const float* __restrict__ b,
                              float* __restrict__ out, int n) {
    int i = blockIdx.x * blockDim.x + threadIdx.x;
    if (i < n * HID_DIM) {
        out[i] = fmaxf(agg[i] + b[i & (HID_DIM - 1)], 0.0f);
    }
}

// ---------------------------------------------------------------------------
// 6. GEMM2: Z[100000,32] = H[100000,64] @ W2[64,32], same WMMA scheme,
//    2 tiles per wave; full K fits in one LDS chunk (stride 68).
// ---------------------------------------------------------------------------
__global__ __launch_bounds__(256) void gcn_gemm2(const float* __restrict__ H,
                                                 const float* __restrict__ W,
                                                 float* __restrict__ Z,
                                                 int nRows) {
    __shared__ float sW[OUT_DIM * 68];           // 8.7 KB, transposed [n][k]
    const int lane = threadIdx.x & 31;
    const int wave = threadIdx.x >> 5;
    const int rowBase = (blockIdx.x * 8 + wave) * 16;
    const bool active = rowBase < nRows;
    const int ln = lane & 15;
    const int kh = lane >> 4;

    for (int i = threadIdx.x; i < HID_DIM * OUT_DIM; i += 256) {
        int k  = i >> 5;      // 0..63
        int nn = i & 31;      // 0..31
        sW[nn * 68 + k] = W[(size_t)k * OUT_DIM + nn];
    }
    __syncthreads();

    if (active) {
        v8f acc0 = {}, acc1 = {};
        const float* hrow = H + (size_t)(rowBase + ln) * HID_DIM;
#pragma unroll
        for (int k0 = 0; k0 < HID_DIM; k0 += 4) {
            const int kb = k0 + kh * 2;
            float2 av = *(const float2*)(hrow + kb);
            v2f a; a[0] = av.x; a[1] = av.y;

            float2 w0 = *(const float2*)(&sW[ln * 68 + kb]);
            float2 w1 = *(const float2*)(&sW[(16 + ln) * 68 + kb]);

            v2f b;
            b[0] = w0.x; b[1] = w0.y;
            acc0 = __builtin_amdgcn_wmma_f32_16x16x4_f32(false, a, false, b,
                                                         (short)0, acc0, false, false);
            b[0] = w1.x; b[1] = w1.y;
            acc1 = __builtin_amdgcn_wmma_f32_16x16x4_f32(false, a, false, b,
                                                         (short)0, acc1, false, false);
        }

        float* zp = Z + (size_t)(rowBase + kh * 8) * OUT_DIM + ln;
#pragma unroll
        for (int j = 0; j < 8; ++j) {
            zp[(size_t)j * OUT_DIM +  0] = acc0[j];
            zp[(size_t)j * OUT_DIM + 16] = acc1[j];
        }
    }
}

// ---------------------------------------------------------------------------
// 7. log_softmax over 32 features: one wave32 per node, lane == feature.
// ---------------------------------------------------------------------------
__global__ void gcn_logsoftmax(float* __restrict__ out,
                               const float* __restrict__ b2, int n) {
    int wid  = (blockIdx.x * blockDim.x + threadIdx.x) >> 5;
    int lane = threadIdx.x & 31;
    if (wid >= n) return;
    float z = out[(size_t)wid * OUT_DIM + lane] + b2[lane];
    float m = z;
#pragma unroll
    for (int off = 16; off > 0; off >>= 1)
        m = fmaxf(m, __shfl_xor(m, off, 32));
    float e = __expf(z - m);
    float s = e;
#pragma unroll
    for (int off = 16; off > 0; off >>= 1)
        s += __shfl_xor(s, off, 32);
    out[(size_t)wid * OUT_DIM + lane] = z - m - __logf(s);
}

// ---------------------------------------------------------------------------
// launch
// ---------------------------------------------------------------------------
extern "C" void kernel_launch(void* const* d_in, const int* in_sizes, int n_in,
                              void* d_out, int out_size, void* d_ws, size_t ws_size,
                              hipStream_t stream) {
    const float* x  = (const float*)d_in[0];
    const int*   ei = (const int*)d_in[1];
    const float* W1 = (const float*)d_in[2];
    const float* b1 = (const float*)d_in[3];
    const float* W2 = (const float*)d_in[4];
    const float* b2 = (const float*)d_in[5];
    float* out = (float*)d_out;

    const int n = in_sizes[0] / IN_DIM;     // 100000 nodes
    const int E = in_sizes[1] / 2;          // 1600000 edges
    const int* row = ei;
    const int* col = ei + E;

    // workspace layout (256B aligned slices)
    char* ws = (char*)d_ws;
    size_t o = 0;
    auto carve = [&](size_t bytes) -> float* {
        float* p = (float*)(ws + o);
        o += (bytes + 255) & ~(size_t)255;
        return p;
    };
    float* dinv = carve((size_t)n * sizeof(float));                 // deg -> dinv
    float* H1   = carve((size_t)n * HID_DIM * sizeof(float));       // gemm1 out, reused for relu out
    float* agg1 = carve((size_t)n * HID_DIM * sizeof(float));       // layer-1 aggregate
    float* Z2   = carve((size_t)n * OUT_DIM * sizeof(float));       // gemm2 out

    const int B = 256;
    const int gemmBlocks = (n + 127) / 128;   // 8 waves x 16 rows per block

    gcn_init<<<(n * HID_DIM + B - 1) / B, B, 0, stream>>>(dinv, agg1, out, n);
    gcn_degree<<<(E + B - 1) / B, B, 0, stream>>>(col, E, dinv);
    gcn_dinv<<<(n + B - 1) / B, B, 0, stream>>>(dinv, n);

    gcn_gemm1<<<gemmBlocks, B, 0, stream>>>(x, W1, H1, n);
    gcn_scatter<HID_DIM><<<(E + n + B - 1) / B, B, 0, stream>>>(row, col, dinv,
                                                                H1, agg1, E, n);
    gcn_bias_relu<<<(n * HID_DIM + B - 1) / B, B, 0, stream>>>(agg1, b1, H1, n);

    gcn_gemm2<<<gemmBlocks, B, 0, stream>>>(H1, W2, Z2, n);
    gcn_scatter<OUT_DIM><<<(E + n + B - 1) / B, B, 0, stream>>>(row, col, dinv,
                                                                Z2, out, E, n);
    gcn_logsoftmax<<<(n * 32 + B - 1) / B, B, 0, stream>>>(out, b2, n);
}